// GCN_unit_4037269258325
// MI455X (gfx1250) — compile-verified
//
#include <hip/hip_runtime.h>

typedef __attribute__((ext_vector_type(2))) float v2f;
typedef __attribute__((ext_vector_type(8))) float v8f;

#define CI   64
#define OO   384
#define VV   27
#define TT   512
#define NN   32
#define CO   128
#define KK   3
#define Tt   4
#define TV   (Tt*VV)        /* 108 */
#define TVP  112            /* padded to 7 tiles of 16 */
#define EPS  1e-5f

/* LDS layout (float offsets) */
#define XS_OFF 0
#define XS_SZ  (CI*TVP)            /* 7168  */
#define YS_OFF (XS_OFF+XS_SZ)
#define YS_SZ  (OO*TVP)            /* 43008 */
#define AS_OFF (YS_OFF+YS_SZ)
#define AS_SZ  (KK*VV*VV)          /* 2187  */
#define SA_OFF (AS_OFF+AS_SZ)
#define SA_SZ  (KK*VV)             /* 81    */
#define RS_OFF (SA_OFF+SA_SZ)      /* per-channel sum   (128) */
#define RQ_OFF (RS_OFF+CO)         /* per-channel sumsq (128) */
#define LDS_FLOATS (RQ_OFF+CO)     /* 52700 floats = ~206 KB  */

__global__ void k_init(float* p) { p[threadIdx.x] = 0.0f; }

__global__ __launch_bounds__(256)
void k_fused(const float* __restrict__ x,  const float* __restrict__ w,
             const float* __restrict__ cb, const float* __restrict__ A,
             float* __restrict__ z, float* __restrict__ gsum, float* __restrict__ gsqs)
{
    extern __shared__ float sm[];
    const int tid = threadIdx.x;
    const int bid = blockIdx.x;
    const int n   = bid >> 7;        /* / (TT/Tt) = /128 */
    const int tb  = bid & 127;
    const int t0  = tb * Tt;
    const float* xg = x + (size_t)n * CI * TT * VV + (size_t)t0 * VV;

    /* ---- stage x into LDS with async DMA (GLOBAL_LOAD_ASYNC_TO_LDS_B128) ---- */
    for (int i = tid; i < CI * (TV / 4); i += 256) {        /* 1728 x float4 */
        const int ci = i / (TV / 4), p4 = i % (TV / 4);
        const float* gp = xg + (size_t)ci * TT * VV + p4 * 4;
        const float* lp = sm + XS_OFF + ci * TVP + p4 * 4;
        const unsigned loff = (unsigned)(uintptr_t)lp;       /* low 32b = LDS offset */
        asm volatile("global_load_async_to_lds_b128 %0, %1, off"
                     :: "v"(loff), "v"(gp) : "memory");
    }
    /* overlap: pad columns, A copy, column sums of A (bias folding) */
    for (int i = tid; i < CI * (TVP - TV); i += 256) {
        int ci = i / (TVP - TV), p = i % (TVP - TV);
        sm[XS_OFF + ci * TVP + TV + p] = 0.0f;
    }
    for (int i = tid; i < AS_SZ; i += 256) sm[AS_OFF + i] = A[i];
    if (tid < SA_SZ) {
        int k = tid / VV, wc = tid % VV;
        float s = 0.0f;
        for (int v = 0; v < VV; ++v) s += A[k * VV * VV + v * VV + wc];
        sm[SA_OFF + tid] = s;
    }
    if (tid < CO) { sm[RS_OFF + tid] = 0.0f; sm[RQ_OFF + tid] = 0.0f; }
    asm volatile("s_wait_asynccnt 0" ::: "memory");
    __syncthreads();

    /* ---- stage 1: y[o, tv] = W[o,ci] * x[ci,tv] via V_WMMA_F32_16X16X4_F32.
       Each wave owns 3 o-tiles; W fragments are hoisted to registers and
       reused across all 7 tv-tiles, so the inner loop is pure LDS + WMMA. ---- */
    const int wave = tid >> 5, lane = tid & 31;
    const int m = lane & 15, half = lane >> 4;
    for (int oi = 0; oi < 3; ++oi) {
        const int o0 = (wave * 3 + oi) * 16;
        /* A-operand: lane m = M (=o row), VGPR j -> K = j + 2*half */
        v2f wr[16];
        const float* wrow = w + (size_t)(o0 + m) * CI + 2 * half;
#pragma unroll
        for (int s = 0; s < 16; ++s) wr[s] = *(const v2f*)(wrow + s * 4);
        for (int nt = 0; nt < 7; ++nt) {
            const int tv0 = nt * 16;
            v8f acc = {};
            /* B-operand: lane m = N (=tv col), VGPR j -> K = j + 2*half */
            const float* xrow = sm + XS_OFF + (2 * half) * TVP + tv0 + m;
#pragma unroll
            for (int s = 0; s < 16; ++s) {
                v2f b;
                b.x = xrow[(s * 4 + 0) * TVP];
                b.y = xrow[(s * 4 + 1) * TVP];
                acc = __builtin_amdgcn_wmma_f32_16x16x4_f32(
                    false, wr[s], false, b, (short)0, acc, false, false);
            }
            /* C/D: VGPR r -> M = r + 8*half ; lane m -> N */
            float* yrow = sm + YS_OFF + (size_t)(o0 + 8 * half) * TVP + tv0 + m;
#pragma unroll
            for (int r = 0; r < 8; ++r) yrow[r * TVP] = acc[r];
        }
    }
    __syncthreads();

    /* ---- stage 2: z[c,t,w] = sum_k sum_v y[k*128+c,t,v] * A[k,v,w] + bias term ---- */
    const float* sA = sm + SA_OFF;
    for (int row = tid; row < CO * Tt; row += 256) {
        const int c = row >> 2, tl = row & 3;
        float accw[VV];
        const float b0 = cb[c], b1 = cb[CO + c], b2 = cb[2 * CO + c];
#pragma unroll
        for (int wv = 0; wv < VV; ++wv)
            accw[wv] = b0 * sA[wv] + b1 * sA[VV + wv] + b2 * sA[2 * VV + wv];
        for (int k = 0; k < KK; ++k) {
            const float* yb = sm + YS_OFF + (size_t)(k * CO + c) * TVP + tl * VV;
            const float* Ab = sm + AS_OFF + k * VV * VV;
            for (int v = 0; v < VV; ++v) {
                const float yv = yb[v];
                const float* Ar = Ab + v * VV;   /* uniform LDS address -> broadcast */
#pragma unroll
                for (int wv = 0; wv < VV; ++wv) accw[wv] = fmaf(yv, Ar[wv], accw[wv]);
            }
        }
        float s1 = 0.0f, s2 = 0.0f;
        float* zp = z + (size_t)n * CO * TT * VV + (size_t)c * TT * VV
                      + (size_t)(t0 + tl) * VV;
#pragma unroll
        for (int wv = 0; wv < VV; ++wv) {
            const float v = accw[wv];
            zp[wv] = v; s1 += v; s2 += v * v;
        }
        __hip_atomic_fetch_add(&sm[RS_OFF + c], s1, __ATOMIC_RELAXED, __HIP_MEMORY_SCOPE_WORKGROUP);
        __hip_atomic_fetch_add(&sm[RQ_OFF + c], s2, __ATOMIC_RELAXED, __HIP_MEMORY_SCOPE_WORKGROUP);
    }
    __syncthreads();
    if (tid < CO) {
        __hip_atomic_fetch_add(&gsum[tid], sm[RS_OFF + tid], __ATOMIC_RELAXED, __HIP_MEMORY_SCOPE_AGENT);
        __hip_atomic_fetch_add(&gsqs[tid], sm[RQ_OFF + tid], __ATOMIC_RELAXED, __HIP_MEMORY_SCOPE_AGENT);
    }
}

__global__ void k_stats(const float* __restrict__ gsum, const float* __restrict__ gsqs,
                        const float* __restrict__ gamma, const float* __restrict__ beta,
                        float* __restrict__ scale, float* __restrict__ shift)
{
    const int c = threadIdx.x;
    if (c < CO) {
        const float inv  = 1.0f / (float)(NN * (size_t)TT * VV);
        const float mean = gsum[c] * inv;
        const float var  = gsqs[c] * inv - mean * mean;
        const float sc   = gamma[c] * rsqrtf(var + EPS);
        scale[c] = sc;
        shift[c] = beta[c] - mean * sc;
    }
}

__global__ __launch_bounds__(256)
void k_apply(float* __restrict__ z, const float* __restrict__ scale,
             const float* __restrict__ shift)
{
    const size_t i4 = (size_t)blockIdx.x * 256 + threadIdx.x;
    const int c = (int)((i4 / (TT * VV / 4)) & (CO - 1));   /* 13824/4 = 3456 */
    float4 v = ((float4*)z)[i4];
    const float sc = scale[c], sh = shift[c];
    v.x = fmaxf(fmaf(v.x, sc, sh), 0.0f);
    v.y = fmaxf(fmaf(v.y, sc, sh), 0.0f);
    v.z = fmaxf(fmaf(v.z, sc, sh), 0.0f);
    v.w = fmaxf(fmaf(v.w, sc, sh), 0.0f);
    ((float4*)z)[i4] = v;
}

extern "C" void kernel_launch(void* const* d_in, const int* in_sizes, int n_in,
                              void* d_out, int out_size, void* d_ws, size_t ws_size,
                              hipStream_t stream)
{
    (void)in_sizes; (void)n_in; (void)out_size; (void)ws_size;
    const float* x     = (const float*)d_in[0];
    /* d_in[1] = len_x (unused, always full length) */
    const float* w     = (const float*)d_in[2];
    const float* cb    = (const float*)d_in[3];
    const float* A     = (const float*)d_in[4];
    const float* gamma = (const float*)d_in[5];
    const float* beta  = (const float*)d_in[6];
    float* ws    = (float*)d_ws;
    float* gsum  = ws;         /* 128 */
    float* gsqs  = ws + 128;   /* 128 */
    float* scale = ws + 256;   /* 128 */
    float* shift = ws + 384;   /* 128 */
    float* z = (float*)d_out;

    k_init<<<1, 256, 0, stream>>>(ws);

    const size_t smem = (size_t)LDS_FLOATS * sizeof(float);
    k_fused<<<dim3(NN * (TT / Tt)), 256, smem, stream>>>(x, w, cb, A, z, gsum, gsqs);

    k_stats<<<1, 128, 0, stream>>>(gsum, gsqs, gamma, beta, scale, shift);

    const int total4 = NN * CO * TT * VV / 4;      /* 14155776 */
    k_apply<<<dim3(total4 / 256), 256, 0, stream>>>(z, scale, shift);
}